// FixedAttention_23905787969987
// MI455X (gfx1250) — compile-verified
//
#include <hip/hip_runtime.h>
#include <math.h>

// ---------------------------------------------------------------------------
// FixedAttention: output[b,s,d] = sum_t attn[s,t] * v[b,t,d];  attn_b = bcast
// attn has rank~17 structure -> per-class column sums + elementwise combine.
// Class sums via chained v_wmma_f32_16x16x4_f32 (indicator A-matrix).
// Staging via GLOBAL_LOAD_ASYNC_TO_LDS_B128 (ASYNCcnt path, no VGPR roundtrip).
// ---------------------------------------------------------------------------

typedef __attribute__((ext_vector_type(2))) float v2f;
typedef __attribute__((ext_vector_type(8))) float v8f;
typedef int v4i __attribute__((vector_size(16)));
typedef __attribute__((address_space(1))) v4i* gptr_b128;
typedef __attribute__((address_space(3))) v4i* lptr_b128;

#define BATCH   64
#define NCLS    16
#define KSHOT   32
#define SEQ     513          // N*K + 1
#define DIM     1024
#define DBLK    64           // d-columns per workgroup
#define LSTR    72           // padded LDS row stride (dwords); 2*72=144 ≡ 16 (mod 64)
                             //  -> WMMA B-fetch: lane halves hit disjoint bank sets
#define NTHR    128          // 4 waves; one 16-wide d-tile per wave

__global__ __launch_bounds__(NTHR)
void fixed_attn_main(const float* __restrict__ v,
                     const float* __restrict__ w,
                     float* __restrict__ out)
{
    __shared__ __align__(16) float vt[SEQ * LSTR];    // 147,744 B : v[b,:,dblk:dblk+64]
    __shared__ __align__(16) float tc[NCLS * LSTR];   //   4,608 B : per-class col sums
    __shared__ __align__(16) float tall[LSTR];        //     288 B : total support sum

    const int blk  = blockIdx.x;
    const int b    = blk >> 4;             // DIM/DBLK = 16 d-slabs
    const int dblk = (blk & 15) * DBLK;
    const int tid  = threadIdx.x;

    const float* vb = v + (size_t)b * SEQ * DIM + dblk;

    // ---- Phase 1: async-stage v slab (513 x 64 f32) into LDS, b128 chunks ----
    for (int ch = tid; ch < SEQ * (DBLK / 4); ch += NTHR) {
        const int t  = ch >> 4;            // row 0..512
        const int c4 = (ch & 15) << 2;     // col 0,4,...,60
        const float* gsrc = vb + (size_t)t * DIM + c4;
        float* ldst = &vt[t * LSTR + c4];
        __builtin_amdgcn_global_load_async_to_lds_b128(
            (gptr_b128)gsrc, (lptr_b128)ldst, /*offset=*/0, /*cpol=*/0);
    }
    asm volatile("s_wait_asynccnt 0x0" ::: "memory");
    __syncthreads();

    // ---- Phase 2: T_cls[16 x 16] per wave via chained WMMA f32 16x16x4 ----
    // A[m,k] = (m == class(t4)) ? 1 : 0   (aligned 4-token groups share class)
    // B[k,n] = v[t4+k, d0+n]
    const int wave = tid >> 5;
    const int lane = tid & 31;
    const int ln   = lane & 15;            // M (A) / N (B) index
    const int hb   = lane >> 4;            // lane-half selects K pairs {0,1} / {2,3}
    const int d0   = wave * 16;

    v8f acc = {0.f, 0.f, 0.f, 0.f, 0.f, 0.f, 0.f, 0.f};

    #pragma unroll 4
    for (int t4 = 0; t4 < SEQ - 1; t4 += 4) {
        const int   cls = t4 >> 5;                     // KSHOT = 32
        const float av  = (ln == cls) ? 1.0f : 0.0f;
        v2f a = {av, av};
        const float* p = &vt[(t4 + (hb << 1)) * LSTR + d0 + ln];
        v2f bop = {p[0], p[LSTR]};
        acc = __builtin_amdgcn_wmma_f32_16x16x4_f32(
            /*neg_a=*/false, a, /*neg_b=*/false, bop,
            /*c_mod=*/(short)0, acc, /*reuse_a=*/false, /*reuse_b=*/false);
    }

    // D layout: VGPR j -> row (j + 8*half), col = lane%16
    #pragma unroll
    for (int j = 0; j < 8; ++j)
        tc[(j + 8 * hb) * LSTR + d0 + ln] = acc[j];
    __syncthreads();

    if (tid < DBLK / 4) {                  // vectorized T_all: 16 threads x float4
        const int c4 = tid << 2;
        float4 s = *reinterpret_cast<const float4*>(&tc[c4]);
        #pragma unroll
        for (int c = 1; c < NCLS; ++c) {
            const float4 t4v = *reinterpret_cast<const float4*>(&tc[c * LSTR + c4]);
            s.x += t4v.x; s.y += t4v.y; s.z += t4v.z; s.w += t4v.w;
        }
        *reinterpret_cast<float4*>(&tall[c4]) = s;
    }
    __syncthreads();

    // ---- Phase 3: elementwise combine, vectorized b128 stream stores ----
    const float w0 = w[0], w1 = w[1], w2 = w[2], w3 = w[3], w4 = w[4], w5 = w[5];
    const float ns = fabsf(w0) + 31.f * fabsf(w1) + 480.f * fabsf(w2) + fabsf(w3);
    const float nq = 512.f * fabsf(w4) + fabsf(w5);
    const float ins = 1.f / ns, inq = 1.f / nq;

    const float cAll  = w2 * ins;
    const float cCls  = (w1 - w2) * ins;
    const float cDiag = (w0 - w1) * ins;
    const float cQcol = w3 * inq;          // support row, t = S-1 column
    const float cQAll = w4 * ins;          // query row, support columns
    const float cQQ   = w5 * inq;          // query row, query column

    float* ob = out + (size_t)b * SEQ * DIM + dblk;
    for (int e = tid; e < SEQ * (DBLK / 4); e += NTHR) {
        const int s  = e >> 4;
        const int c4 = (e & 15) << 2;
        const float4 vq = *reinterpret_cast<const float4*>(&vt[(SEQ - 1) * LSTR + c4]);
        const float4 ta = *reinterpret_cast<const float4*>(&tall[c4]);
        float4 res;
        if (s < SEQ - 1) {
            const float4 tcv = *reinterpret_cast<const float4*>(&tc[(s >> 5) * LSTR + c4]);
            const float4 vv  = *reinterpret_cast<const float4*>(&vt[s * LSTR + c4]);
            res.x = cAll * ta.x + cCls * tcv.x + cDiag * vv.x + cQcol * vq.x;
            res.y = cAll * ta.y + cCls * tcv.y + cDiag * vv.y + cQcol * vq.y;
            res.z = cAll * ta.z + cCls * tcv.z + cDiag * vv.z + cQcol * vq.z;
            res.w = cAll * ta.w + cCls * tcv.w + cDiag * vv.w + cQcol * vq.w;
        } else {
            res.x = cQAll * ta.x + cQQ * vq.x;
            res.y = cQAll * ta.y + cQQ * vq.y;
            res.z = cQAll * ta.z + cQQ * vq.z;
            res.w = cQAll * ta.w + cQQ * vq.w;
        }
        *reinterpret_cast<float4*>(ob + (size_t)s * DIM + c4) = res;
    }
}

// ---- attn_b: the 513x513 matrix (6 distinct values), broadcast over batch ----
__global__ __launch_bounds__(256)
void fixed_attn_mat(const float* __restrict__ w, float* __restrict__ attn_out)
{
    const int idx = blockIdx.x * 256 + threadIdx.x;
    if (idx >= SEQ * SEQ) return;
    const int s = idx / SEQ;
    const int t = idx - s * SEQ;

    const float w0 = w[0], w1 = w[1], w2 = w[2], w3 = w[3], w4 = w[4], w5 = w[5];
    const float ns = fabsf(w0) + 31.f * fabsf(w1) + 480.f * fabsf(w2) + fabsf(w3);
    const float nq = 512.f * fabsf(w4) + fabsf(w5);
    const float ins = 1.f / ns, inq = 1.f / nq;

    float val;
    if (s == SEQ - 1)              val = (t == SEQ - 1) ? w5 * inq : w4 * ins;
    else if (t == SEQ - 1)         val = w3 * inq;
    else if (t == s)               val = w0 * ins;
    else if ((t >> 5) == (s >> 5)) val = w1 * ins;
    else                           val = w2 * ins;

    #pragma unroll 4
    for (int b = 0; b < BATCH; ++b)
        attn_out[(size_t)b * (SEQ * SEQ) + idx] = val;
}

extern "C" void kernel_launch(void* const* d_in, const int* in_sizes, int n_in,
                              void* d_out, int out_size, void* d_ws, size_t ws_size,
                              hipStream_t stream)
{
    // inputs: q, k, v, attn_weight, N, K  (q/k unused by the reference output)
    const float* v = (const float*)d_in[2];
    const float* w = (const float*)d_in[3];
    float* out      = (float*)d_out;
    float* attn_out = out + (size_t)BATCH * SEQ * DIM;

    fixed_attn_main<<<dim3(BATCH * (DIM / DBLK)), dim3(NTHR), 0, stream>>>(v, w, out);

    const int nelem = SEQ * SEQ;
    fixed_attn_mat<<<dim3((nelem + 255) / 256), dim3(256), 0, stream>>>(w, attn_out);
}